// GMMLayer_47725676593246
// MI455X (gfx1250) — compile-verified
//
#include <hip/hip_runtime.h>

typedef __attribute__((ext_vector_type(2))) float v2f;
typedef __attribute__((ext_vector_type(8))) float v8f;

#define N_NODES 50000
#define IN_DIM  128
#define OUT_DIM 128
#define KKER    3
#define COLS    (KKER * OUT_DIM)   /* 384 */
#define BN_EPS  1e-5f
#define SLOPE   0.01f

// ---------------------------------------------------------------------------
// Kernel 0: agg[n, c] = bias[c]  (bias folded in before scatter-add);
// zero the BN moment accumulators. Runs every call (ws is not re-poisoned).
// ---------------------------------------------------------------------------
__global__ __launch_bounds__(256)
void init_agg(float* __restrict__ agg, float* __restrict__ sums,
              float* __restrict__ sumsq, const float* __restrict__ bias) {
    int i = blockIdx.x * 256 + threadIdx.x;
    if (i < N_NODES * OUT_DIM) agg[i] = bias[i & (OUT_DIM - 1)];
    if (i < OUT_DIM) { sums[i] = 0.0f; sumsq[i] = 0.0f; }
}

// ---------------------------------------------------------------------------
// Kernel 1: proj[N, 384] = h[N,128] @ fc_w[128,384] via V_WMMA_F32_16X16X4_F32
// One wave (32 lanes) computes one 16x16 tile, looping K in steps of 4.
// A 16x4 f32 layout: lanes 0-15 carry M=lane with K=0,1 in the 2 VGPRs;
// lanes 16-31 carry K=2,3.  B 4x16: N=lane&15, same K split.
// 3125 * 24 = 75000 tiles, 8 waves/block -> 9375 blocks, no remainder,
// so EXEC is all-ones at every WMMA (wave-uniform exit only).
// ---------------------------------------------------------------------------
__global__ __launch_bounds__(256)
void gemm_proj(const float* __restrict__ h, const float* __restrict__ w,
               float* __restrict__ proj) {
    const int lane   = threadIdx.x;                 // 0..31
    const int tile   = blockIdx.x * 8 + threadIdx.y;
    const int ntiles = COLS / 16;                   // 24
    const int mt     = tile / ntiles;
    const int nt     = tile % ntiles;

    const int lan15  = lane & 15;
    const int khalf  = (lane >> 4) * 2;             // 0 or 2
    const int row    = mt * 16 + lan15;
    const int col    = nt * 16 + lan15;

    const float* __restrict__ arow = h + (size_t)row * IN_DIM;

    v8f acc = {};
    #pragma unroll 4
    for (int kk = 0; kk < IN_DIM; kk += 4) {
        v2f a;
        a.x = arow[kk + khalf];
        a.y = arow[kk + khalf + 1];
        v2f b;
        b.x = w[(size_t)(kk + khalf) * COLS + col];
        b.y = w[(size_t)(kk + khalf + 1) * COLS + col];
        // (neg_a, A, neg_b, B, c_mod, C, reuse_a, reuse_b)
        acc = __builtin_amdgcn_wmma_f32_16x16x4_f32(
            false, a, false, b, (short)0, acc, false, false);
    }

    // C/D 16x16 f32 layout: VGPR r -> M = r (lanes 0-15) or M = 8+r (16-31)
    const int rbase = mt * 16 + ((lane >> 4) << 3);
    float* __restrict__ outp = proj + col;
    #pragma unroll
    for (int r = 0; r < 8; ++r)
        outp[(size_t)(rbase + r) * COLS] = acc[r];
}

// ---------------------------------------------------------------------------
// Kernel 2: per-edge Gaussian mix + gather proj[src] + scatter-add to agg[dst].
// One wave per edge; each lane owns 4 channels (float4 = b128 loads).
// proj (76.8 MB) and agg (25.6 MB) are L2-resident on MI455X (192 MB L2),
// so the 1.23 GB gather + 0.41 GB atomic traffic stays on-die.
// ---------------------------------------------------------------------------
__global__ __launch_bounds__(256)
void edge_scatter(const float* __restrict__ proj, const float* __restrict__ pseudo,
                  const int* __restrict__ src, const int* __restrict__ dst,
                  const float* __restrict__ mu, const float* __restrict__ invs,
                  float* __restrict__ agg, int n_edges) {
    const int e = blockIdx.x * 8 + threadIdx.y;     // wave-uniform
    if (e >= n_edges) return;
    const int lane = threadIdx.x;

    const int s = src[e];
    const int d = dst[e];
    const float p0 = pseudo[2 * e];
    const float p1 = pseudo[2 * e + 1];

    float g[KKER];
    #pragma unroll
    for (int k = 0; k < KKER; ++k) {
        float d0 = (p0 - mu[2 * k])     * invs[2 * k];
        float d1 = (p1 - mu[2 * k + 1]) * invs[2 * k + 1];
        g[k] = expf(-0.5f * (d0 * d0 + d1 * d1));
    }

    const float* __restrict__ prow = proj + (size_t)s * COLS + lane * 4;
    float4 m0 = *(const float4*)(prow);
    float4 m1 = *(const float4*)(prow + OUT_DIM);
    float4 m2 = *(const float4*)(prow + 2 * OUT_DIM);

    float4 msg;
    msg.x = g[0] * m0.x + g[1] * m1.x + g[2] * m2.x;
    msg.y = g[0] * m0.y + g[1] * m1.y + g[2] * m2.y;
    msg.z = g[0] * m0.z + g[1] * m1.z + g[2] * m2.z;
    msg.w = g[0] * m0.w + g[1] * m1.w + g[2] * m2.w;

    float* __restrict__ ar = agg + (size_t)d * OUT_DIM + lane * 4;
    atomicAdd(ar + 0, msg.x);
    atomicAdd(ar + 1, msg.y);
    atomicAdd(ar + 2, msg.z);
    atomicAdd(ar + 3, msg.w);
}

// ---------------------------------------------------------------------------
// Kernel 3: per-channel moments. Thread = channel, rows strided over blocks,
// register accumulation then 2 f32 atomics per (block, channel).
// ---------------------------------------------------------------------------
__global__ __launch_bounds__(128)
void bn_stats(const float* __restrict__ agg, float* __restrict__ sums,
              float* __restrict__ sumsq) {
    const int c = threadIdx.x;
    float s = 0.0f, q = 0.0f;
    for (int r = blockIdx.x; r < N_NODES; r += gridDim.x) {
        float v = agg[(size_t)r * OUT_DIM + c];
        s += v;
        q += v * v;
    }
    atomicAdd(&sums[c], s);
    atomicAdd(&sumsq[c], q);
}

// ---------------------------------------------------------------------------
// Kernel 4: BN (training-mode, biased var) + leaky_relu + residual.
// ---------------------------------------------------------------------------
__global__ __launch_bounds__(256)
void bn_final(const float* __restrict__ agg, const float* __restrict__ sums,
              const float* __restrict__ sumsq, const float* __restrict__ gamma,
              const float* __restrict__ beta, const float* __restrict__ h,
              float* __restrict__ out) {
    int i = blockIdx.x * 256 + threadIdx.x;
    if (i >= N_NODES * OUT_DIM) return;
    const int c = i & (OUT_DIM - 1);
    const float inv_n = 1.0f / (float)N_NODES;
    float mean = sums[c] * inv_n;
    float var  = sumsq[c] * inv_n - mean * mean;
    float x = (agg[i] - mean) * rsqrtf(var + BN_EPS) * gamma[c] + beta[c];
    x = x > 0.0f ? x : SLOPE * x;
    out[i] = x + h[i];
}

// ---------------------------------------------------------------------------
extern "C" void kernel_launch(void* const* d_in, const int* in_sizes, int n_in,
                              void* d_out, int out_size, void* d_ws, size_t ws_size,
                              hipStream_t stream) {
    const float* h      = (const float*)d_in[0];
    const float* pseudo = (const float*)d_in[1];
    const int*   src    = (const int*)d_in[2];
    const int*   dst    = (const int*)d_in[3];
    const float* fc_w   = (const float*)d_in[4];
    const float* mu     = (const float*)d_in[5];
    const float* invs   = (const float*)d_in[6];
    const float* bias   = (const float*)d_in[7];
    const float* gamma  = (const float*)d_in[8];
    const float* beta   = (const float*)d_in[9];
    float*       out    = (float*)d_out;
    const int n_edges   = in_sizes[2];

    // Workspace layout: proj (76.8 MB) | agg (25.6 MB) | sums | sumsq
    float* proj  = (float*)d_ws;
    float* agg   = proj + (size_t)N_NODES * COLS;
    float* sums  = agg + (size_t)N_NODES * OUT_DIM;
    float* sumsq = sums + OUT_DIM;

    const int tot = N_NODES * OUT_DIM;
    dim3 wblk(32, 8);   // 8 waves of 32 (wave32)

    init_agg<<<(tot + 255) / 256, 256, 0, stream>>>(agg, sums, sumsq, bias);

    const int ntile_total = (N_NODES / 16) * (COLS / 16);   // 75000, /8 exact
    gemm_proj<<<ntile_total / 8, wblk, 0, stream>>>(h, fc_w, proj);

    edge_scatter<<<(n_edges + 7) / 8, wblk, 0, stream>>>(
        proj, pseudo, src, dst, mu, invs, agg, n_edges);

    bn_stats<<<512, 128, 0, stream>>>(agg, sums, sumsq);

    bn_final<<<(tot + 255) / 256, 256, 0, stream>>>(
        agg, sums, sumsq, gamma, beta, h, out);
}